// GNNEncoder_3092376453137
// MI455X (gfx1250) — compile-verified
//
#include <hip/hip_runtime.h>

#define N_NODES 50000
#define N_EDGES 800000
#define IN_DIM  64
#define HID_DIM 128
#define OUT_DIM 64
#define BN_EPS  1e-5f

typedef float v2f __attribute__((ext_vector_type(2)));
typedef float v8f __attribute__((ext_vector_type(8)));

// ---------------------------------------------------------------- utilities
__global__ void zero_f32(float* __restrict__ p, int n) {
    int i = blockIdx.x * blockDim.x + threadIdx.x;
    if (i < n) p[i] = 0.0f;
}

__global__ void deg_count(const int* __restrict__ dst, float* __restrict__ deg, int E) {
    int e = blockIdx.x * blockDim.x + threadIdx.x;
    if (e < E) unsafeAtomicAdd(&deg[dst[e]], 1.0f);
}

__global__ void deg_invert(float* __restrict__ deg, int n) {
    int i = blockIdx.x * blockDim.x + threadIdx.x;
    if (i < n) { float d = deg[i]; deg[i] = (d > 0.0f) ? 1.0f / d : 0.0f; }
}

// ------------------------------------------------- edge scatter (L2-resident)
// One thread per (edge, feature). Feature dim D is a power of two, block=256.
template <int D>
__global__ void scatter_add(const float* __restrict__ feat,
                            const int* __restrict__ src,
                            const int* __restrict__ dst,
                            float* __restrict__ agg, int E) {
    const int epb = 256 / D;                       // edges per block
    int e = blockIdx.x * epb + (threadIdx.x / D);
    int d = threadIdx.x & (D - 1);
    if (e < E) {
        int s = src[e];
        int t = dst[e];
        unsafeAtomicAdd(&agg[(size_t)t * D + d], feat[(size_t)s * D + d]);
    }
}

// ------------------------------------------------- fused SAGEConv GEMM (WMMA)
// out[m, n] = sum_k (agg[m,k] * dinv[m]) * Wl[k,n]  +  sum_k self[m,k] * Wr[k,n]  + bias[n]
// One wave32 computes one 16x16 tile with V_WMMA_F32_16X16X4_F32 (exact fp32).
// blockDim.x = (N/16) * 32; gridDim.x = M/16 (M = 50000 is a multiple of 16).
template <int K, int N>
__global__ void sage_gemm(const float* __restrict__ Aagg,   // [M,K]
                          const float* __restrict__ Aself,  // [M,K]
                          const float* __restrict__ dinvv,  // [M]
                          const float* __restrict__ Wl,     // [K,N]
                          const float* __restrict__ Wr,     // [K,N]
                          const float* __restrict__ bias,   // [N]
                          float* __restrict__ outp) {       // [M,N]
    const int lane = threadIdx.x & 31;
    const int wave = threadIdx.x >> 5;          // n-tile handled by this wave
    const int row0 = blockIdx.x * 16;
    const int col0 = wave * 16;

    const int ln16 = lane & 15;                 // A-row / B-col / D-col in tile
    const int half = lane >> 4;                 // 0: K=0,1  1: K=2,3 sub-slice
    const int kb   = half * 2;

    const int   arow    = row0 + ln16;
    const float dinv    = dinvv[arow];
    const float* aggRow  = Aagg  + (size_t)arow * K;
    const float* selfRow = Aself + (size_t)arow * K;

    v8f c = {};                                 // fp32 accumulator, 8 VGPRs
#pragma unroll 8
    for (int k0 = 0; k0 < K; k0 += 4) {
        // --- neighbor-aggregate term: (agg * deg_inv) @ Wl
        v2f a = *reinterpret_cast<const v2f*>(aggRow + k0 + kb);   // 8B load
        a.x *= dinv; a.y *= dinv;
        v2f b;
        b.x = Wl[(size_t)(k0 + kb)     * N + col0 + ln16];
        b.y = Wl[(size_t)(k0 + kb + 1) * N + col0 + ln16];
        c = __builtin_amdgcn_wmma_f32_16x16x4_f32(false, a, false, b,
                                                  (short)0, c, false, false);
        // --- root term: self @ Wr
        v2f a2 = *reinterpret_cast<const v2f*>(selfRow + k0 + kb); // 8B load
        v2f b2;
        b2.x = Wr[(size_t)(k0 + kb)     * N + col0 + ln16];
        b2.y = Wr[(size_t)(k0 + kb + 1) * N + col0 + ln16];
        c = __builtin_amdgcn_wmma_f32_16x16x4_f32(false, a2, false, b2,
                                                  (short)0, c, false, false);
    }

    const float bc = bias[col0 + ln16];
#pragma unroll
    for (int v = 0; v < 8; ++v) {
        int row = row0 + v + 8 * half;          // C/D layout: VGPR v -> M = v + 8*half
        outp[(size_t)row * N + col0 + ln16] = c[v] + bc;
    }
}

// ------------------------------------------------------------- batch norm
// Column-parallel partial sums (coalesced: thread t owns column t), fp atomics
// into stats[0:128]=sum, stats[128:256]=sumsq.
__global__ void bn_stats(const float* __restrict__ h, float* __restrict__ stats, int M) {
    int col = threadIdx.x;                       // 0..127
    float s = 0.0f, q = 0.0f;
    for (int r = blockIdx.x; r < M; r += gridDim.x) {
        float v = h[(size_t)r * HID_DIM + col];
        s += v; q += v * v;
    }
    unsafeAtomicAdd(&stats[col], s);
    unsafeAtomicAdd(&stats[HID_DIM + col], q);
}

// Fold mean/var/gamma/beta into per-column scale & shift.
__global__ void bn_finalize(const float* __restrict__ stats,
                            const float* __restrict__ gamma,
                            const float* __restrict__ beta,
                            float* __restrict__ scale,
                            float* __restrict__ shift, float invM) {
    int c = threadIdx.x;
    float mu   = stats[c] * invM;
    float var  = stats[HID_DIM + c] * invM - mu * mu;
    float rstd = rsqrtf(var + BN_EPS);
    float sc   = gamma[c] * rstd;
    scale[c] = sc;
    shift[c] = beta[c] - mu * sc;
}

__global__ void bn_relu(float* __restrict__ h,
                        const float* __restrict__ scale,
                        const float* __restrict__ shift, int total) {
    int i = blockIdx.x * blockDim.x + threadIdx.x;
    if (i < total) {
        int col = i & (HID_DIM - 1);
        float v = h[i] * scale[col] + shift[col];
        h[i] = v > 0.0f ? v : 0.0f;
    }
}

// ---------------------------------------------------------------- launcher
extern "C" void kernel_launch(void* const* d_in, const int* in_sizes, int n_in,
                              void* d_out, int out_size, void* d_ws, size_t ws_size,
                              hipStream_t stream) {
    (void)in_sizes; (void)n_in; (void)out_size; (void)ws_size;

    const float* x     = (const float*)d_in[0];
    const int*   ei    = (const int*)  d_in[1];   // [2, N_EDGES]
    const float* W1l   = (const float*)d_in[2];
    const float* b1    = (const float*)d_in[3];
    const float* W1r   = (const float*)d_in[4];
    const float* gamma = (const float*)d_in[5];
    const float* beta  = (const float*)d_in[6];
    const float* W2l   = (const float*)d_in[7];
    const float* b2    = (const float*)d_in[8];
    const float* W2r   = (const float*)d_in[9];
    float*       out   = (float*)d_out;

    const int* src = ei;             // edge_index[0]
    const int* dst = ei + N_EDGES;   // edge_index[1]

    // workspace layout (floats)
    float* ws    = (float*)d_ws;
    float* deg   = ws;                                  // N_NODES (also deg_inv)
    float* agg   = ws + 50176;                          // N_NODES*HID_DIM (shared by L1/L2)
    float* h     = agg + (size_t)N_NODES * HID_DIM;     // N_NODES*HID_DIM
    float* stats = h   + (size_t)N_NODES * HID_DIM;     // 512: sum, sumsq, scale, shift

    const int THR = 256;

    // degrees -> deg_inv (in place)
    zero_f32<<<(N_NODES + THR - 1) / THR, THR, 0, stream>>>(deg, N_NODES);
    deg_count<<<(N_EDGES + THR - 1) / THR, THR, 0, stream>>>(dst, deg, N_EDGES);
    deg_invert<<<(N_NODES + THR - 1) / THR, THR, 0, stream>>>(deg, N_NODES);

    // layer 1: scatter-mean + fused GEMM  (h_pre -> h buffer)
    zero_f32<<<(N_NODES * IN_DIM + THR - 1) / THR, THR, 0, stream>>>(agg, N_NODES * IN_DIM);
    scatter_add<IN_DIM><<<(N_EDGES + 3) / 4, THR, 0, stream>>>(x, src, dst, agg, N_EDGES);
    sage_gemm<IN_DIM, HID_DIM><<<N_NODES / 16, (HID_DIM / 16) * 32, 0, stream>>>(
        agg, x, deg, W1l, W1r, b1, h);

    // batch norm (training stats) + ReLU, in place on h
    zero_f32<<<1, 2 * HID_DIM, 0, stream>>>(stats, 2 * HID_DIM);
    bn_stats<<<512, HID_DIM, 0, stream>>>(h, stats, N_NODES);
    bn_finalize<<<1, HID_DIM, 0, stream>>>(stats, gamma, beta,
                                           stats + 2 * HID_DIM, stats + 3 * HID_DIM,
                                           1.0f / (float)N_NODES);
    bn_relu<<<(N_NODES * HID_DIM + THR - 1) / THR, THR, 0, stream>>>(
        h, stats + 2 * HID_DIM, stats + 3 * HID_DIM, N_NODES * HID_DIM);

    // layer 2: scatter-mean + fused GEMM -> out
    zero_f32<<<(N_NODES * HID_DIM + THR - 1) / THR, THR, 0, stream>>>(agg, N_NODES * HID_DIM);
    scatter_add<HID_DIM><<<(N_EDGES + 1) / 2, THR, 0, stream>>>(h, src, dst, agg, N_EDGES);
    sage_gemm<HID_DIM, OUT_DIM><<<N_NODES / 16, (OUT_DIM / 16) * 32, 0, stream>>>(
        agg, h, deg, W2l, W2r, b2, out);
}